// CanadarmJacob_58265526337792
// MI455X (gfx1250) — compile-verified
//
#include <hip/hip_runtime.h>
#include <hip/hip_bf16.h>
#include <stdint.h>

// Problem geometry (fixed by the reference)
#define N_PAIRS (512 * 256)   // N_S * N_H independent problems
#define LP_F    144           // 4*4*9 floats of link_pose per pair
#define COM_F   21            // 3*7 floats of com per pair
#define OUT_F   42            // 6*7 floats of output per pair
#define PPB     64            // pairs per block
#define TPB     64            // threads per block (2 wave32)

typedef __attribute__((ext_vector_type(4))) unsigned u32x4;
typedef __attribute__((ext_vector_type(8))) unsigned u32x8;

// Generic pointers to LDS carry the wave-relative LDS byte offset in the low 32 bits.
__device__ __forceinline__ unsigned lds_off(const void* p) {
  return (unsigned)(uintptr_t)p;
}

// ---- ASYNCcnt path: async global<->LDS b128 with compile-time inst offsets ----
// INST_OFFSET is applied to BOTH the LDS and the global address (ISA 15.18),
// so a strided bulk copy needs only one base VGPR pair + one LDS-offset VGPR.
template <int J, int N, int STRIDE>
struct AsyncLd {
  static __device__ __forceinline__ void run(unsigned lds, const void* g) {
    asm volatile("global_load_async_to_lds_b128 %0, %1, off offset:%2"
                 :: "v"(lds), "v"(g), "i"(J * STRIDE) : "memory");
    AsyncLd<J + 1, N, STRIDE>::run(lds, g);
  }
};
template <int N, int STRIDE>
struct AsyncLd<N, N, STRIDE> {
  static __device__ __forceinline__ void run(unsigned, const void*) {}
};

template <int J, int N, int STRIDE>
struct AsyncSt {
  static __device__ __forceinline__ void run(void* g, unsigned lds) {
    asm volatile("global_store_async_from_lds_b128 %0, %1, off offset:%2"
                 :: "v"(g), "v"(lds), "i"(J * STRIDE) : "memory");
    AsyncSt<J + 1, N, STRIDE>::run(g, lds);
  }
};
template <int N, int STRIDE>
struct AsyncSt<N, N, STRIDE> {
  static __device__ __forceinline__ void run(void*, unsigned) {}
};

__device__ __forceinline__ void wait_async0() {
  asm volatile("s_wait_asynccnt 0x0" ::: "memory");
}

// ---- TENSORcnt path: one TDM descriptor copies a whole block's LP chunk ----
// D# group0/group1 per CDNA5 ISA 8.3/8.4 (2D form, VADDR2/3 disabled).
__device__ __forceinline__ void tdm_load_1d(unsigned lds_addr, const void* gaddr,
                                            unsigned nelem /* 4-byte elements */) {
  const unsigned long long ga = (unsigned long long)(uintptr_t)gaddr;
  u32x4 g0;
  g0[0] = 1u;                                   // count=1, is_restore=0, gather=0
  g0[1] = lds_addr;                             // lds_addr (bytes)
  g0[2] = (unsigned)(ga & 0xFFFFFFFFull);       // global_addr[31:0]
  g0[3] = ((unsigned)(ga >> 32) & 0x01FFFFFFu)  // global_addr[56:32]
          | 0x80000000u;                        // type=2 ("image") in [127:126]
  u32x8 g1;
  g1[0] = 0x00020000u;                          // data_size=2 (4 bytes), no multicast
  g1[1] = (nelem & 0xFFFFu) << 16;              // tensor_dim0[15:0]
  g1[2] = ((nelem >> 16) & 0xFFFFu)             // tensor_dim0[31:16]
          | (1u << 16);                         // tensor_dim1 = 1
  g1[3] = (nelem & 0xFFFFu) << 16;              // tile_dim0 = nelem (16-bit)
  g1[4] = 1u;                                   // tile_dim1 = 1, tile_dim2 = 0
  g1[5] = nelem;                                // tensor_dim0_stride[31:0]
  g1[6] = ((nelem >> 16) & 0xFFFFu)             // tensor_dim0_stride[47:32]
          | ((nelem & 0xFFFFu) << 16);          // tensor_dim1_stride[15:0]
  g1[7] = (nelem >> 16) & 0xFFFFu;              // tensor_dim1_stride[47:16]
  asm volatile("tensor_load_to_lds %0, %1" :: "s"(g0), "s"(g1) : "memory");
}

__device__ __forceinline__ void cross3(float u0, float u1, float u2,
                                       float v0, float v1, float v2,
                                       float& r0, float& r1, float& r2) {
  r0 = u1 * v2 - u2 * v1;
  r1 = u2 * v0 - u0 * v2;
  r2 = u0 * v1 - u1 * v0;
}

__global__ __launch_bounds__(TPB)
void canadarm_jacob_kernel(const float* __restrict__ com_g,
                           const float* __restrict__ lp_g,
                           const int* __restrict__ bm_g,
                           float* __restrict__ out_g) {
  __shared__ __attribute__((aligned(16))) float sLP[PPB * LP_F];    // 36,864 B
  __shared__ __attribute__((aligned(16))) float sCOM[PPB * COM_F];  //  5,376 B
  __shared__ __attribute__((aligned(16))) float sOUT[PPB * OUT_F];  // 10,752 B

  const int tid = threadIdx.x;
  const long long p0 = (long long)blockIdx.x * PPB;
  const float* gLP  = lp_g  + p0 * LP_F;
  const float* gCM  = com_g + p0 * COM_F;
  float*       gOUT = out_g + p0 * OUT_F;

  // ---- Stage link_pose chunk (36,864 B contiguous) via one TDM transfer ----
  if (tid < 32) {                                   // one wave issues the DMA
    tdm_load_1d(lds_off(&sLP[0]), gLP, PPB * LP_F); // 9216 x 4B elements
  }

  // ---- Stage com chunk via ASYNCcnt b128 loads (336 float4: 5 full + 16) ----
  const float* gCMb = gCM + tid * 4;
  const unsigned lCM = lds_off(&sCOM[tid * 4]);
  AsyncLd<0, 5, TPB * 16>::run(lCM, gCMb);          // offsets 0..4*1024
  if (tid < 16) AsyncLd<5, 6, TPB * 16>::run(lCM, gCMb);

  wait_async0();
  __builtin_amdgcn_s_wait_tensorcnt(0);             // no-op for non-issuing waves
  __syncthreads();

  // ---- Compile-time model constants ----
  const int   AX[7] = {2, 0, 2, 2, 2, 0, 2};
  const float SG[7] = {1.f, 1.f, 1.f, 1.f, -1.f, 1.f, 1.f};
  const float MS[7] = {105.98f, 105.98f, 314.98f, 279.2f, 105.98f, 105.98f, 243.66f};
  const float DG[7][3] = {
      {12.19f, 12.19f, 3.061f},   {12.19f, 12.19f, 3.061f},
      {15.41f, 2094.71f, 2103.19f}, {9.522f, 1966.28f, 1966.28f},
      {8.305f, 3.061f, 8.0386f},  {12.13f, 12.13f, 3.061f},
      {9.336f, 44.41f, 44.41f}};
  float SMS[7];       // suffix sums of MASS
  float SDG[7][3];    // suffix sums of inertia diagonals (SUM_INERTIAL is diagonal)
  SMS[6] = MS[6];
#pragma unroll
  for (int a = 5; a >= 0; --a) SMS[a] = SMS[a + 1] + MS[a];
#pragma unroll
  for (int i = 0; i < 3; ++i) SDG[6][i] = DG[6][i];
#pragma unroll
  for (int a = 5; a >= 0; --a)
#pragma unroll
    for (int i = 0; i < 3; ++i) SDG[a][i] = SDG[a + 1][i] + DG[a][i];
  const float TM  = 100000.0f + 243.66f + SMS[0];                    // TOTAL_MASS
  const float BCZ = 6.65f * 243.66f / (100000.0f + 243.66f);         // BASE_COM.z

  // ---- Per-thread gather of one record from LDS ----
  const float* L = &sLP[tid * LP_F];
  const float* C = &sCOM[tid * COM_F];
  float rot[3][7], pos[3][8], cm[3][7];
#pragma unroll
  for (int r = 0; r < 3; ++r) {
#pragma unroll
    for (int a = 0; a < 7; ++a) {
      rot[r][a] = L[(r * 4 + AX[a]) * 9 + a] * SG[a];
      cm[r][a]  = C[r * 7 + a];
    }
#pragma unroll
    for (int l = 0; l < 8; ++l) pos[r][l] = L[(r * 4 + 3) * 9 + l];
  }

  float o[6][7];
  const int bm = bm_g[0];
  if (bm) {
    // del_p, jacob_t, jacob_tw
    float dp[3][7], jt[3][7], jtw[3][7];
#pragma unroll
    for (int a = 0; a < 7; ++a) {
#pragma unroll
      for (int r = 0; r < 3; ++r) dp[r][a] = cm[r][a] - pos[r][a];
      cross3(rot[0][a], rot[1][a], rot[2][a], dp[0][a], dp[1][a], dp[2][a],
             jt[0][a], jt[1][a], jt[2][a]);
#pragma unroll
      for (int r = 0; r < 3; ++r) jtw[r][a] = jt[r][a] * SMS[a];
    }

    // H_w = diag(SDG[0]) + sum_a m_a * (|dp|^2 I - dp dp^T)   (symmetric)
    float Hw00 = SDG[0][0], Hw11 = SDG[0][1], Hw22 = SDG[0][2];
    float Hw01 = 0.f, Hw02 = 0.f, Hw12 = 0.f;
#pragma unroll
    for (int a = 0; a < 7; ++a) {
      const float x = dp[0][a], y = dp[1][a], z = dp[2][a];
      const float r2 = x * x + y * y + z * z;
      const float m = MS[a];
      Hw00 += m * (r2 - x * x);
      Hw11 += m * (r2 - y * y);
      Hw22 += m * (r2 - z * z);
      Hw01 -= m * x * y;
      Hw02 -= m * x * z;
      Hw12 -= m * y * z;
    }

    // Suffix-weighted position sums: term2[:,k] = w_k x jt[:,k]
    float w[3][7];
#pragma unroll
    for (int r = 0; r < 3; ++r) w[r][6] = MS[6] * dp[r][6];
#pragma unroll
    for (int a = 5; a >= 0; --a)
#pragma unroll
      for (int r = 0; r < 3; ++r) w[r][a] = w[r][a + 1] + MS[a] * dp[r][a];

    // H_wphi = diag-scale(rot) + w x jt
    float Hph[3][7];
#pragma unroll
    for (int a = 0; a < 7; ++a) {
      float c0, c1, c2;
      cross3(w[0][a], w[1][a], w[2][a], jt[0][a], jt[1][a], jt[2][a], c0, c1, c2);
      Hph[0][a] = SDG[a][0] * rot[0][a] + c0;
      Hph[1][a] = SDG[a][1] * rot[1][a] + c1;
      Hph[2][a] = SDG[a][2] * rot[2][a] + c2;
    }

    // q = system_com - BASE_COM
    float q0 = 0.f, q1 = 0.f, q2 = 0.f;
#pragma unroll
    for (int a = 0; a < 7; ++a) {
      q0 += MS[a] * cm[0][a];
      q1 += MS[a] * cm[1][a];
      q2 += MS[a] * cm[2][a];
    }
    q0 /= TM; q1 /= TM; q2 = q2 / TM - BCZ;

    // H_s = TM*(q q^T - |q|^2 I) + H_w    (r_og @ r_og expanded)
    const float qq = q0 * q0 + q1 * q1 + q2 * q2;
    const float a_ = TM * (q0 * q0 - qq) + Hw00;
    const float b_ = TM * (q0 * q1) + Hw01;
    const float c_ = TM * (q0 * q2) + Hw02;
    const float d_ = b_;
    const float e_ = TM * (q1 * q1 - qq) + Hw11;
    const float f_ = TM * (q1 * q2) + Hw12;
    const float g_ = c_;
    const float h_ = f_;
    const float i_ = TM * (q2 * q2 - qq) + Hw22;

    // inv3 via adjugate (matches reference)
    const float det = a_ * (e_ * i_ - f_ * h_) - b_ * (d_ * i_ - f_ * g_) +
                      c_ * (d_ * h_ - e_ * g_);
    const float id = 1.0f / det;
    float Hi[3][3];
    Hi[0][0] =  (e_ * i_ - f_ * h_) * id;
    Hi[0][1] = -(b_ * i_ - c_ * h_) * id;
    Hi[0][2] =  (b_ * f_ - c_ * e_) * id;
    Hi[1][0] = -(d_ * i_ - f_ * g_) * id;
    Hi[1][1] =  (a_ * i_ - c_ * g_) * id;
    Hi[1][2] = -(a_ * f_ - c_ * d_) * id;
    Hi[2][0] =  (d_ * h_ - e_ * g_) * id;
    Hi[2][1] = -(a_ * h_ - b_ * g_) * id;
    Hi[2][2] =  (a_ * e_ - b_ * d_) * id;

    const float invTM = 1.0f / TM;
#pragma unroll
    for (int a = 0; a < 7; ++a) {
      // H_theta[:,a] = Hph[:,a] - q x jtw[:,a]
      float cx, cy, cz;
      cross3(q0, q1, q2, jtw[0][a], jtw[1][a], jtw[2][a], cx, cy, cz);
      const float h0 = Hph[0][a] - cx;
      const float h1 = Hph[1][a] - cy;
      const float h2 = Hph[2][a] - cz;
      const float M0 = Hi[0][0] * h0 + Hi[0][1] * h1 + Hi[0][2] * h2;
      const float M1 = Hi[1][0] * h0 + Hi[1][1] * h1 + Hi[1][2] * h2;
      const float M2 = Hi[2][0] * h0 + Hi[2][1] * h1 + Hi[2][2] * h2;
      float rx, ry, rz;
      cross3(q0, q1, q2, M0, M1, M2, rx, ry, rz);
      o[0][a] = -(jtw[0][a] * invTM + rx);
      o[1][a] = -(jtw[1][a] * invTM + ry);
      o[2][a] = -(jtw[2][a] * invTM + rz);
      o[3][a] = -M0;
      o[4][a] = -M1;
      o[5][a] = -M2;
    }
  } else {
    // bm == 0: del_p from link 7 (index -2 of 9); out = [cross(rot,dp); rot]
#pragma unroll
    for (int a = 0; a < 7; ++a) {
      const float dx = pos[0][7] - pos[0][a];
      const float dy = pos[1][7] - pos[1][a];
      const float dz = pos[2][7] - pos[2][a];
      float c0, c1, c2;
      cross3(rot[0][a], rot[1][a], rot[2][a], dx, dy, dz, c0, c1, c2);
      o[0][a] = c0; o[1][a] = c1; o[2][a] = c2;
      o[3][a] = rot[0][a]; o[4][a] = rot[1][a]; o[5][a] = rot[2][a];
    }
  }

  // ---- Stage result in LDS, then coalesced async b128 stores from LDS ----
#pragma unroll
  for (int r = 0; r < 6; ++r)
#pragma unroll
    for (int a = 0; a < 7; ++a) sOUT[tid * OUT_F + r * 7 + a] = o[r][a];
  __syncthreads();

  float* gOUTb = gOUT + tid * 4;                    // 672 float4: 10 full + 32
  const unsigned lOUT = lds_off(&sOUT[tid * 4]);
  AsyncSt<0, 10, TPB * 16>::run(gOUTb, lOUT);       // offsets 0..9*1024
  if (tid < 32) AsyncSt<10, 11, TPB * 16>::run(gOUTb, lOUT);
  wait_async0();
}

extern "C" void kernel_launch(void* const* d_in, const int* in_sizes, int n_in,
                              void* d_out, int out_size, void* d_ws, size_t ws_size,
                              hipStream_t stream) {
  const float* com = (const float*)d_in[0];   // (512,256,3,7)
  const float* lp  = (const float*)d_in[1];   // (512,256,4,4,9)
  const int*   bm  = (const int*)d_in[2];     // scalar
  float*       out = (float*)d_out;           // (512,256,6,7)
  (void)in_sizes; (void)n_in; (void)out_size; (void)d_ws; (void)ws_size;

  const int blocks = N_PAIRS / PPB;           // 2048
  hipLaunchKernelGGL(canadarm_jacob_kernel, dim3(blocks), dim3(TPB), 0, stream,
                     com, lp, bm, out);
}